// GhostTraceGNN_66503273611267
// MI455X (gfx1250) — compile-verified
//
#include <hip/hip_runtime.h>
#include <hip/hip_bf16.h>
#include <math.h>

#define N_NODES 50000
#define N_EDGES 800000
#define E_TOT   (N_EDGES + N_NODES)   // self-loops appended
#define N_GRAPHS 16

typedef __attribute__((ext_vector_type(2))) float v2f;
typedef __attribute__((ext_vector_type(8))) float v8f;

static __device__ __forceinline__ void atomicMaxF(float* addr, float v) {
  // IEEE-754 monotonic trick: positives compare as signed int, negatives reversed as unsigned
  if (v >= 0.0f) atomicMax((int*)addr, __float_as_int(v));
  else           atomicMin((unsigned int*)addr, __float_as_uint(v));
}

static __device__ __forceinline__ float lrelu(float v) {
  return (v > 0.0f) ? v : 0.2f * v;
}

// ---------------- padding helpers (layer 1: K=14 -> 16) ----------------
__global__ void pad_x_kernel(const float* __restrict__ x, float* __restrict__ xp) {
  int idx = blockIdx.x * blockDim.x + threadIdx.x;
  if (idx >= N_NODES * 16) return;
  int n = idx >> 4, k = idx & 15;
  xp[idx] = (k < 14) ? x[n * 14 + k] : 0.0f;
}

__global__ void pad_w_kernel(const float* __restrict__ w, float* __restrict__ wp,
                             int Ksrc, int N) {
  int idx = blockIdx.x * blockDim.x + threadIdx.x;
  if (idx >= 16 * N) return;
  int k = idx / N, n = idx % N;
  wp[idx] = (k < Ksrc) ? w[k * N + n] : 0.0f;
}

// ---------------- f32 WMMA GEMM: C[M,N] = A[M,K] @ B[K,N] ----------------
// One wave per 16x64 output tile: 4 x V_WMMA_F32_16X16X4_F32 per K-step sharing
// one A-fragment load (4x A reuse). N must be a multiple of 64.
// A 16x4 layout: lanes 0-15 hold M=lane,K={k0,k0+1}; lanes 16-31 hold K={k0+2,k0+3}.
// B 4x16 layout: lanes 0-15 hold N=lane,K={k0,k0+1}; lanes 16-31 hold K={k0+2,k0+3}.
// C/D: VGPR r -> (M = r + 8*hi, N = lane&15).
__global__ void wmma_gemm_f32(const float* __restrict__ A, const float* __restrict__ B,
                              float* __restrict__ C, int M, int N, int K) {
  int wid  = (blockIdx.x * blockDim.x + threadIdx.x) >> 5;
  int lane = threadIdx.x & 31;
  int ntile = N >> 6;                 // 64-wide tiles
  int tiles = (M >> 4) * ntile;
  if (wid >= tiles) return;
  int mt = wid / ntile, nt = wid % ntile;
  int col = lane & 15;   // M-row for A, N-col for B/C
  int hi  = lane >> 4;   // K sub-pair select

  const float* arow = A + (size_t)(mt * 16 + col) * K + 2 * hi;
  v8f acc0 = (v8f)0.0f, acc1 = (v8f)0.0f, acc2 = (v8f)0.0f, acc3 = (v8f)0.0f;
  for (int k0 = 0; k0 < K; k0 += 4) {
    v2f a;
    a.x = arow[k0];
    a.y = arow[k0 + 1];
    const float* bp = B + (size_t)(k0 + 2 * hi) * N + nt * 64 + col;
    v2f b0, b1, b2, b3;
    b0.x = bp[0];  b0.y = bp[N];
    b1.x = bp[16]; b1.y = bp[N + 16];
    b2.x = bp[32]; b2.y = bp[N + 32];
    b3.x = bp[48]; b3.y = bp[N + 48];
    acc0 = __builtin_amdgcn_wmma_f32_16x16x4_f32(false, a, false, b0, (short)0, acc0, false, false);
    acc1 = __builtin_amdgcn_wmma_f32_16x16x4_f32(false, a, false, b1, (short)0, acc1, false, false);
    acc2 = __builtin_amdgcn_wmma_f32_16x16x4_f32(false, a, false, b2, (short)0, acc2, false, false);
    acc3 = __builtin_amdgcn_wmma_f32_16x16x4_f32(false, a, false, b3, (short)0, acc3, false, false);
  }
  float* cp = C + (size_t)(mt * 16 + hi * 8) * N + nt * 64 + col;
#pragma unroll
  for (int r = 0; r < 8; ++r) {
    cp[(size_t)r * N]      = acc0[r];
    cp[(size_t)r * N + 16] = acc1[r];
    cp[(size_t)r * N + 32] = acc2[r];
    cp[(size_t)r * N + 48] = acc3[r];
  }
}

// ---------------- GAT edge stages ----------------
__global__ void init_seg_kernel(float* __restrict__ emax, float* __restrict__ den, int n) {
  int idx = blockIdx.x * blockDim.x + threadIdx.x;
  if (idx >= n) return;
  emax[idx] = __int_as_float(0xff800000);  // -inf
  den[idx]  = 0.0f;
}

__global__ void init_out_bias_kernel(float* __restrict__ out, const float* __restrict__ bias,
                                     int total, int F) {
  int idx = blockIdx.x * blockDim.x + threadIdx.x;
  if (idx >= total) return;
  out[idx] = bias[idx % F];
}

// Pass A: e[edge,h] = sum_c lrelu(xl[src]+xr[dst]) * att[h,c] ; atomic max into emax[dst,h]
// Wave per 4 edges; b128 loads: lane owns 4 (H=2) or 2 (H=1) contiguous features.
__global__ void edge_logits_kernel(const float* __restrict__ xl, const float* __restrict__ xr,
                                   const float* __restrict__ att, const int* __restrict__ ei,
                                   float* __restrict__ ebuf, float* __restrict__ emax, int H) {
  const int F = H * 64;
  int wid  = (blockIdx.x * blockDim.x + threadIdx.x) >> 5;
  int lane = threadIdx.x & 31;
  // per-lane attention fragment (uniform across edges)
  float4 a4; float2 a2;
  if (H == 2) a4 = ((const float4*)att)[lane];
  else        a2 = ((const float2*)att)[lane];
  int e0 = wid * 4;
  for (int e = e0; e < e0 + 4 && e < E_TOT; ++e) {
    int s, d;
    if (e < N_EDGES) { s = ei[e]; d = ei[N_EDGES + e]; } else { s = d = e - N_EDGES; }
    if (e + 1 < e0 + 4 && e + 1 < E_TOT) {  // prefetch next edge's rows (L2-resident)
      int s2, d2;
      if (e + 1 < N_EDGES) { s2 = ei[e + 1]; d2 = ei[N_EDGES + e + 1]; }
      else                 { s2 = d2 = e + 1 - N_EDGES; }
      __builtin_prefetch(xl + (size_t)s2 * F + lane * (F / 32), 0, 1);
      __builtin_prefetch(xr + (size_t)d2 * F + lane * (F / 32), 0, 1);
    }
    if (H == 2) {
      // lanes 0-15 own head 0 (feat 4*lane..), lanes 16-31 own head 1
      float4 xa = ((const float4*)(xl + (size_t)s * 128))[lane];
      float4 xb = ((const float4*)(xr + (size_t)d * 128))[lane];
      float p = lrelu(xa.x + xb.x) * a4.x + lrelu(xa.y + xb.y) * a4.y +
                lrelu(xa.z + xb.z) * a4.z + lrelu(xa.w + xb.w) * a4.w;
      for (int o = 8; o; o >>= 1) p += __shfl_xor(p, o, 32);  // 16-lane butterfly per head
      if ((lane & 15) == 0) {
        int h = lane >> 4;
        ebuf[(size_t)e * 2 + h] = p;
        atomicMaxF(&emax[(size_t)d * 2 + h], p);
      }
    } else {
      float2 xa = ((const float2*)(xl + (size_t)s * 64))[lane];
      float2 xb = ((const float2*)(xr + (size_t)d * 64))[lane];
      float p = lrelu(xa.x + xb.x) * a2.x + lrelu(xa.y + xb.y) * a2.y;
      for (int o = 16; o; o >>= 1) p += __shfl_xor(p, o, 32);
      if (lane == 0) {
        ebuf[e] = p;
        atomicMaxF(&emax[d], p);
      }
    }
  }
}

// Pass B: denom[dst,h] += exp(e - emax[dst,h])
__global__ void edge_denom_kernel(const float* __restrict__ ebuf, const float* __restrict__ emax,
                                  float* __restrict__ den, const int* __restrict__ ei, int H) {
  int idx = blockIdx.x * blockDim.x + threadIdx.x;
  if (idx >= E_TOT * H) return;
  int e = idx / H, h = idx % H;
  int d = (e < N_EDGES) ? ei[N_EDGES + e] : e - N_EDGES;
  float p = __expf(ebuf[idx] - emax[(size_t)d * H + h]);
  atomicAdd(&den[(size_t)d * H + h], p);
}

// Pass C: out[dst] += xl[src] * alpha  (alpha recomputed; out pre-initialized to bias)
__global__ void edge_scatter_kernel(const float* __restrict__ xl, const float* __restrict__ ebuf,
                                    const float* __restrict__ emax, const float* __restrict__ den,
                                    const int* __restrict__ ei, float* __restrict__ out, int H) {
  const int F = H * 64;
  int wid  = (blockIdx.x * blockDim.x + threadIdx.x) >> 5;
  int lane = threadIdx.x & 31;
  int e0 = wid * 4;
  for (int e = e0; e < e0 + 4 && e < E_TOT; ++e) {
    int s, d;
    if (e < N_EDGES) { s = ei[e]; d = ei[N_EDGES + e]; } else { s = d = e - N_EDGES; }
    if (e + 1 < e0 + 4 && e + 1 < E_TOT) {
      int s2 = (e + 1 < N_EDGES) ? ei[e + 1] : e + 1 - N_EDGES;
      __builtin_prefetch(xl + (size_t)s2 * F + lane * (F / 32), 0, 1);
    }
    if (H == 2) {
      int h = lane >> 4;  // lanes 0-15: head 0, lanes 16-31: head 1
      float alpha = __expf(ebuf[(size_t)e * 2 + h] - emax[(size_t)d * 2 + h]) /
                    den[(size_t)d * 2 + h];
      float4 v = ((const float4*)(xl + (size_t)s * 128))[lane];
      float* op = out + (size_t)d * 128 + lane * 4;
      atomicAdd(op + 0, v.x * alpha);
      atomicAdd(op + 1, v.y * alpha);
      atomicAdd(op + 2, v.z * alpha);
      atomicAdd(op + 3, v.w * alpha);
    } else {
      float alpha = __expf(ebuf[e] - emax[d]) / den[d];
      float2 v = ((const float2*)(xl + (size_t)s * 64))[lane];
      float* op = out + (size_t)d * 64 + lane * 2;
      atomicAdd(op + 0, v.x * alpha);
      atomicAdd(op + 1, v.y * alpha);
    }
  }
}

// ---------------- LayerNorm + ELU + residual/skip epilogue (wave per node) ----------------
__global__ void post_ln_kernel(const float* __restrict__ gin, const float* __restrict__ g,
                               const float* __restrict__ be, const float* __restrict__ skip,
                               const float* __restrict__ sbias, const float* __restrict__ resid,
                               float* __restrict__ hout, float* __restrict__ xin_out,
                               int N, int F) {
  int wid  = (blockIdx.x * blockDim.x + threadIdx.x) >> 5;
  int lane = threadIdx.x & 31;
  if (wid >= N) return;
  if (F == 128) {
    float4 v = ((const float4*)(gin + (size_t)wid * 128))[lane];
    float s = v.x + v.y + v.z + v.w;
    for (int o = 16; o; o >>= 1) s += __shfl_xor(s, o, 32);
    float mu = s * (1.0f / 128.0f);
    float dx = v.x - mu, dy = v.y - mu, dz = v.z - mu, dw = v.w - mu;
    float q = dx * dx + dy * dy + dz * dz + dw * dw;
    for (int o = 16; o; o >>= 1) q += __shfl_xor(q, o, 32);
    float rstd = rsqrtf(q * (1.0f / 128.0f) + 1e-5f);
    float4 gg = ((const float4*)g)[lane];
    float4 bb = ((const float4*)be)[lane];
    float4 y;
    y.x = dx * rstd * gg.x + bb.x; y.y = dy * rstd * gg.y + bb.y;
    y.z = dz * rstd * gg.z + bb.z; y.w = dw * rstd * gg.w + bb.w;
    y.x = (y.x > 0.0f) ? y.x : (__expf(y.x) - 1.0f);
    y.y = (y.y > 0.0f) ? y.y : (__expf(y.y) - 1.0f);
    y.z = (y.z > 0.0f) ? y.z : (__expf(y.z) - 1.0f);
    y.w = (y.w > 0.0f) ? y.w : (__expf(y.w) - 1.0f);
    if (skip) {
      float4 sk = ((const float4*)(skip + (size_t)wid * 128))[lane];
      y.x += sk.x; y.y += sk.y; y.z += sk.z; y.w += sk.w;
    }
    if (sbias) {
      float4 sb = ((const float4*)sbias)[lane];
      y.x += sb.x; y.y += sb.y; y.z += sb.z; y.w += sb.w;
    }
    if (resid) {
      float4 rs = ((const float4*)(resid + (size_t)wid * 128))[lane];
      y.x += rs.x; y.y += rs.y; y.z += rs.z; y.w += rs.w;
    }
    ((float4*)(hout + (size_t)wid * 128))[lane] = y;
    if (xin_out) ((float4*)(xin_out + (size_t)wid * 128))[lane] = y;
  } else {  // F == 64
    float2 v = ((const float2*)(gin + (size_t)wid * 64))[lane];
    float s = v.x + v.y;
    for (int o = 16; o; o >>= 1) s += __shfl_xor(s, o, 32);
    float mu = s * (1.0f / 64.0f);
    float dx = v.x - mu, dy = v.y - mu;
    float q = dx * dx + dy * dy;
    for (int o = 16; o; o >>= 1) q += __shfl_xor(q, o, 32);
    float rstd = rsqrtf(q * (1.0f / 64.0f) + 1e-5f);
    float2 gg = ((const float2*)g)[lane];
    float2 bb = ((const float2*)be)[lane];
    float2 y;
    y.x = dx * rstd * gg.x + bb.x; y.y = dy * rstd * gg.y + bb.y;
    y.x = (y.x > 0.0f) ? y.x : (__expf(y.x) - 1.0f);
    y.y = (y.y > 0.0f) ? y.y : (__expf(y.y) - 1.0f);
    if (skip) {
      float2 sk = ((const float2*)(skip + (size_t)wid * 64))[lane];
      y.x += sk.x; y.y += sk.y;
    }
    if (sbias) {
      float2 sb = ((const float2*)sbias)[lane];
      y.x += sb.x; y.y += sb.y;
    }
    if (resid) {
      float2 rs = ((const float2*)(resid + (size_t)wid * 64))[lane];
      y.x += rs.x; y.y += rs.y;
    }
    ((float2*)(hout + (size_t)wid * 64))[lane] = y;
    if (xin_out) ((float2*)(xin_out + (size_t)wid * 64))[lane] = y;
  }
}

// ---------------- graph pooling (mean + max over 16 graphs, F=64) ----------------
__global__ void pool_init_kernel(float* psum, float* pmax, float* pcnt) {
  int idx = blockIdx.x * blockDim.x + threadIdx.x;
  if (idx < N_GRAPHS * 64) { psum[idx] = 0.0f; pmax[idx] = __int_as_float(0xff800000); }
  if (idx < N_GRAPHS) pcnt[idx] = 0.0f;
}

__global__ void pool_accum_kernel(const float* __restrict__ h, const int* __restrict__ batch,
                                  float* psum, float* pmax, float* pcnt) {
  int wid  = (blockIdx.x * blockDim.x + threadIdx.x) >> 5;
  int lane = threadIdx.x & 31;
  if (wid >= N_NODES) return;
  int gidx = batch[wid];
  float2 v = ((const float2*)(h + (size_t)wid * 64))[lane];
  float* ps = psum + gidx * 64 + lane * 2;
  float* pm = pmax + gidx * 64 + lane * 2;
  atomicAdd(ps + 0, v.x);
  atomicAdd(ps + 1, v.y);
  atomicMaxF(pm + 0, v.x);
  atomicMaxF(pm + 1, v.y);
  if (lane == 0) atomicAdd(&pcnt[gidx], 1.0f);
}

__global__ void pool_final_kernel(const float* psum, const float* pmax, const float* pcnt,
                                  float* __restrict__ out) {
  int idx = blockIdx.x * blockDim.x + threadIdx.x;
  if (idx >= N_GRAPHS * 64) return;
  int gidx = idx >> 6, c = idx & 63;
  float cnt = pcnt[gidx];
  cnt = (cnt > 1.0f) ? cnt : 1.0f;
  out[gidx * 128 + c]      = psum[idx] / cnt;   // pooled_mean
  out[gidx * 128 + 64 + c] = pmax[idx];         // pooled_max
}

// ---------------- host orchestration ----------------
static inline int divup(int a, int b) { return (a + b - 1) / b; }

extern "C" void kernel_launch(void* const* d_in, const int* in_sizes, int n_in,
                              void* d_out, int out_size, void* d_ws, size_t ws_size,
                              hipStream_t stream) {
  const float* x   = (const float*)d_in[0];
  const int*   ei  = (const int*)d_in[1];     // [2, 800000] flattened
  const int*   bat = (const int*)d_in[2];
  const float* Wl1 = (const float*)d_in[3];
  const float* Wr1 = (const float*)d_in[4];
  const float* a1  = (const float*)d_in[5];
  const float* b1  = (const float*)d_in[6];
  const float* g1  = (const float*)d_in[7];
  const float* be1 = (const float*)d_in[8];
  const float* Ws1 = (const float*)d_in[9];
  const float* bs1 = (const float*)d_in[10];
  const float* Wl2 = (const float*)d_in[11];
  const float* Wr2 = (const float*)d_in[12];
  const float* a2  = (const float*)d_in[13];
  const float* b2  = (const float*)d_in[14];
  const float* g2  = (const float*)d_in[15];
  const float* be2 = (const float*)d_in[16];
  const float* Wl3 = (const float*)d_in[17];
  const float* Wr3 = (const float*)d_in[18];
  const float* a3  = (const float*)d_in[19];
  const float* b3  = (const float*)d_in[20];
  const float* g3  = (const float*)d_in[21];
  const float* be3 = (const float*)d_in[22];
  const float* Ws3 = (const float*)d_in[23];
  const float* bs3 = (const float*)d_in[24];

  float* ws = (float*)d_ws;
  size_t off = 0;
  auto alloc = [&](size_t n) { float* p = ws + off; off += n; return p; };
  float* xpad = alloc((size_t)N_NODES * 16);
  float* wl1p = alloc(16 * 128);
  float* wr1p = alloc(16 * 128);
  float* ws1p = alloc(16 * 128);
  float* xl   = alloc((size_t)N_NODES * 128);
  float* xr   = alloc((size_t)N_NODES * 128);
  float* skip = alloc((size_t)N_NODES * 128);
  float* gout = alloc((size_t)N_NODES * 128);
  float* hbuf = alloc((size_t)N_NODES * 128);
  float* xin  = alloc((size_t)N_NODES * 128);
  float* ebuf = alloc((size_t)E_TOT * 2);
  float* emax = alloc((size_t)N_NODES * 2);
  float* den  = alloc((size_t)N_NODES * 2);
  float* psum = alloc(N_GRAPHS * 64);
  float* pmax = alloc(N_GRAPHS * 64);
  float* pcnt = alloc(N_GRAPHS);

  auto gemm = [&](const float* A, const float* B, float* C, int M, int N, int K) {
    int tiles = (M / 16) * (N / 64);   // 16x64 tile per wave
    wmma_gemm_f32<<<divup(tiles, 8), 256, 0, stream>>>(A, B, C, M, N, K);
  };
  int ewaves = divup(E_TOT, 4);
  auto gat = [&](const float* XL, const float* XR, const float* att, const float* bias,
                 int H, float* OUT) {
    int F = H * 64;
    init_seg_kernel<<<divup(N_NODES * H, 256), 256, 0, stream>>>(emax, den, N_NODES * H);
    init_out_bias_kernel<<<divup(N_NODES * F, 256), 256, 0, stream>>>(OUT, bias, N_NODES * F, F);
    edge_logits_kernel<<<divup(ewaves, 8), 256, 0, stream>>>(XL, XR, att, ei, ebuf, emax, H);
    edge_denom_kernel<<<divup(E_TOT * H, 256), 256, 0, stream>>>(ebuf, emax, den, ei, H);
    edge_scatter_kernel<<<divup(ewaves, 8), 256, 0, stream>>>(XL, ebuf, emax, den, ei, OUT, H);
  };

  // ---- Layer 1 (14 -> 128, H=2) ----
  pad_x_kernel<<<divup(N_NODES * 16, 256), 256, 0, stream>>>(x, xpad);
  pad_w_kernel<<<divup(16 * 128, 256), 256, 0, stream>>>(Wl1, wl1p, 14, 128);
  pad_w_kernel<<<divup(16 * 128, 256), 256, 0, stream>>>(Wr1, wr1p, 14, 128);
  pad_w_kernel<<<divup(16 * 128, 256), 256, 0, stream>>>(Ws1, ws1p, 14, 128);
  gemm(xpad, wl1p, xl, N_NODES, 128, 16);
  gemm(xpad, wr1p, xr, N_NODES, 128, 16);
  gemm(xpad, ws1p, skip, N_NODES, 128, 16);
  gat(xl, xr, a1, b1, 2, gout);
  post_ln_kernel<<<divup(N_NODES, 8), 256, 0, stream>>>(
      gout, g1, be1, skip, bs1, nullptr, hbuf, xin, N_NODES, 128);

  // ---- Layer 2 (128 -> 128, H=2, identity residual) ----
  gemm(hbuf, Wl2, xl, N_NODES, 128, 128);
  gemm(hbuf, Wr2, xr, N_NODES, 128, 128);
  gat(xl, xr, a2, b2, 2, gout);
  post_ln_kernel<<<divup(N_NODES, 8), 256, 0, stream>>>(
      gout, g2, be2, nullptr, nullptr, xin, hbuf, xin, N_NODES, 128);

  // ---- Layer 3 (128 -> 64, H=1, skip GEMM) ----
  gemm(hbuf, Wl3, xl, N_NODES, 64, 128);
  gemm(hbuf, Wr3, xr, N_NODES, 64, 128);
  gemm(hbuf, Ws3, skip, N_NODES, 64, 128);
  gat(xl, xr, a3, b3, 1, gout);
  post_ln_kernel<<<divup(N_NODES, 8), 256, 0, stream>>>(
      gout, g3, be3, skip, bs3, nullptr, hbuf, nullptr, N_NODES, 64);

  // ---- Pooling -> [16, 128] ----
  pool_init_kernel<<<divup(N_GRAPHS * 64, 256), 256, 0, stream>>>(psum, pmax, pcnt);
  pool_accum_kernel<<<divup(N_NODES, 8), 256, 0, stream>>>(hbuf, bat, psum, pmax, pcnt);
  pool_final_kernel<<<divup(N_GRAPHS * 64, 256), 256, 0, stream>>>(psum, pmax, pcnt,
                                                                   (float*)d_out);
  (void)in_sizes; (void)n_in; (void)out_size; (void)ws_size;
}